// mLinReg_22969485099688
// MI455X (gfx1250) — compile-verified
//
#include <hip/hip_runtime.h>

// Reference: out[e,n,o] = sum_t inp[e,t,n] * W[o,t] + b[o]
//   E=32, T=256, N=8192, O=256, all fp32.
//
// Per-epoch GEMM via V_WMMA_F32_16X16X4_F32 (fp32 in/out, K=4 per instr).
// Memory-bound (AI ~64 FLOP/B; ~537MB @ 23.3TB/s ~= 23us floor), so the fp32
// matrix path keeps reference precision at no roofline cost.
//
// W is staged in LDS K-pair interleaved: Wp[t/2][o] = {W[o][t], W[o][t+1]},
// so each 4x16 B fragment is ONE ds_load_b64 straight into an even-aligned
// VGPR pair (no cross-register shuffles before v_wmma).

typedef __attribute__((ext_vector_type(2))) float v2f;
typedef __attribute__((ext_vector_type(8))) float v8f;

#define T_DIM 256
#define N_DIM 8192
#define O_DIM 256
// Row stride in K-pairs: 272 pairs = 544 dwords ≡ 32 (mod 64 banks) ->
// the two half-waves of a B-fragment b64 load cover disjoint 32-bank halves.
#define PAIR_STRIDE 272
#define LDS_BYTES ((T_DIM / 2) * PAIR_STRIDE * 8)   // 278528 B < 320KB WGP LDS

__global__ __launch_bounds__(512) void mLinReg_wmma_f32_kernel(
    const float* __restrict__ inp,   // [E, T, N]
    const float* __restrict__ W,     // [O, T]
    const float* __restrict__ bias,  // [O]
    float* __restrict__ out)         // [E, N, O]
{
  extern __shared__ v2f Wp[];        // Wp[(t/2) * PAIR_STRIDE + o] = {W[o,t], W[o,t+1]}
  const int tid = threadIdx.x;
  const int e   = blockIdx.y;
  const int n0  = blockIdx.x * 64;

  // ---- Stage W transposed+pair-interleaved into LDS (256KB payload) ----
  {
    const int o  = tid & 255;          // one W row per thread (2 threads/row)
    const int th = (tid >> 8) * 128;   // which half of the t range
    const float4* W4 = reinterpret_cast<const float4*>(W + o * T_DIM + th);
    #pragma unroll 4
    for (int i = 0; i < 32; ++i) {
      float4 w = W4[i];
      int p = (th >> 1) + i * 2;       // K-pair row
      v2f p0; p0[0] = w.x; p0[1] = w.y;
      v2f p1; p1[0] = w.z; p1[1] = w.w;
      Wp[(p + 0) * PAIR_STRIDE + o] = p0;  // lane-contiguous b64 stores:
      Wp[(p + 1) * PAIR_STRIDE + o] = p1;  // all 64 banks hit once
    }
  }
  __syncthreads();

  // ---- Wave tiling: 16 waves = 4 (n) x 4 (o); wave tile = 16n x 64o ----
  const int lane = tid & 31;
  const int wave = tid >> 5;
  const int wn   = wave >> 2;
  const int wo   = wave & 3;
  const int row  = lane & 15;           // M index within 16x16
  const int kp   = lane >> 4;           // K half: lanes<16 -> pair 0, >=16 -> pair 1
  const int o_l  = wo * 64 + row;       // this lane's o column for B fragments

  // A fragment base: inputs[e, 2*kp, n0 + wn*16 + row]
  const float* A = inp + (size_t)e * T_DIM * N_DIM
                       + (size_t)(kp << 1) * N_DIM + n0 + wn * 16 + row;
  // B fragment base: Wp[tb/2 + kp][o_l + j*16]
  const v2f* B = Wp + (size_t)kp * PAIR_STRIDE + o_l;

  v8f acc[4] = {};   // 4 x (16x16 f32) accumulators = wave's 16x64 tile

  #pragma unroll 4
  for (int tb = 0; tb < T_DIM; tb += 4) {
    // A 16x4 f32 fragment: VGPR0 = K=2kp, VGPR1 = K=2kp+1 (per ISA layout)
    v2f a;
    a[0] = A[(size_t)tb * N_DIM];
    a[1] = A[(size_t)(tb + 1) * N_DIM];
    if (tb + 16 < T_DIM)  // stream ahead: global_prefetch_b8
      __builtin_prefetch(A + (size_t)(tb + 16) * N_DIM, 0, 3);

    const v2f* Bt = B + (size_t)(tb >> 1) * PAIR_STRIDE;
    #pragma unroll
    for (int j = 0; j < 4; ++j) {
      v2f bf = Bt[j * 16];   // one ds_load_b64, even-aligned pair
      acc[j] = __builtin_amdgcn_wmma_f32_16x16x4_f32(
          /*neg_a=*/false, a, /*neg_b=*/false, bf,
          /*c_mod=*/(short)0, acc[j], /*reuse_a=*/false, /*reuse_b=*/false);
    }
  }

  // ---- Epilogue: +bias, store. D layout: VGPR r, lanes<16 -> M=r, N=lane;
  // lanes>=16 -> M=8+r, N=lane-16. Lane-contiguous 64B segments per store. ----
  const int nrow0 = n0 + wn * 16 + (lane >> 4) * 8;
  const int ocol0 = wo * 64 + (lane & 15);
  float bv[4];
  #pragma unroll
  for (int j = 0; j < 4; ++j) bv[j] = bias[ocol0 + j * 16];

  float* outp = out + ((size_t)e * N_DIM + nrow0) * O_DIM + ocol0;
  #pragma unroll
  for (int r = 0; r < 8; ++r) {
    #pragma unroll
    for (int j = 0; j < 4; ++j)
      outp[(size_t)r * O_DIM + j * 16] = acc[j][r] + bv[j];
  }
}

extern "C" void kernel_launch(void* const* d_in, const int* in_sizes, int n_in,
                              void* d_out, int out_size, void* d_ws, size_t ws_size,
                              hipStream_t stream) {
  const float* inp  = (const float*)d_in[0];  // [32, 256, 8192]
  const float* W    = (const float*)d_in[1];  // [256, 256]
  const float* bias = (const float*)d_in[2];  // [256]
  float* out        = (float*)d_out;          // [32, 8192, 256]

  // Allow >64KB (default cap) of dynamic LDS; not a stream op, capture-safe.
  (void)hipFuncSetAttribute((const void*)mLinReg_wmma_f32_kernel,
                            hipFuncAttributeMaxDynamicSharedMemorySize,
                            LDS_BYTES);

  dim3 grid(N_DIM / 64, 32);   // (n-blocks, epochs)
  dim3 block(512);             // 16 waves
  mLinReg_wmma_f32_kernel<<<grid, block, LDS_BYTES, stream>>>(inp, W, bias, out);
}